// GNNLayer_16492674417056
// MI455X (gfx1250) — compile-verified
//
#include <hip/hip_runtime.h>
#include <hip/hip_bf16.h>
#include <math.h>

typedef float v2f __attribute__((ext_vector_type(2)));
typedef float v8f __attribute__((ext_vector_type(8)));

#define DFEAT 64

// ---------------- init: agg (=d_out) to 0, deg to 1 (self loop) ----------------
__global__ __launch_bounds__(256) void gcn_init(float* __restrict__ agg,
                                                float* __restrict__ deg, int N) {
  int j = blockIdx.x * 256 + threadIdx.x;
  if (j < N * DFEAT) agg[j] = 0.0f;
  if (j < N) deg[j] = 1.0f;
}

// ---------------- xw = x @ W via V_WMMA_F32_16X16X4_F32 ----------------
// One wave produces a 16x64 strip of xw. W cached in LDS.
// A (16x4 f32): lanes 0-15 -> M=0..15; VGPR0 = K=(2*half), VGPR1 = K=(2*half+1).
// B (4x16 f32): VGPR0 lanes hold row K=(2*half), N=lane&15; VGPR1 = K+1.
// D (16x16 f32): VGPR v -> M = v + 8*half, N = lane&15.
__global__ __launch_bounds__(256) void gcn_gemm_wmma(const float* __restrict__ x,
                                                     const float* __restrict__ W,
                                                     float* __restrict__ xw, int N) {
  __shared__ float Ws[DFEAT * DFEAT];
  for (int i = threadIdx.x; i < DFEAT * DFEAT; i += 256) Ws[i] = W[i];
  __syncthreads();

  const int wave = threadIdx.x >> 5;
  const int lane = threadIdx.x & 31;
  const int half = lane >> 4;
  const int l16  = lane & 15;
  const int row0 = (blockIdx.x * 8 + wave) * 16;
  if (row0 >= N) return;

  int rr = row0 + l16;
  if (rr >= N) rr = N - 1;                   // clamp (N is a multiple of 16 here)
  const float* __restrict__ xrow = x + (size_t)rr * DFEAT;

  v8f acc[4] = {v8f{}, v8f{}, v8f{}, v8f{}};
#pragma unroll
  for (int k = 0; k < DFEAT; k += 4) {
    const int ka = k + 2 * half;
    v2f a;
    a.x = xrow[ka];
    a.y = xrow[ka + 1];
#pragma unroll
    for (int nt = 0; nt < 4; ++nt) {
      v2f b;
      b.x = Ws[ka * DFEAT + nt * 16 + l16];
      b.y = Ws[(ka + 1) * DFEAT + nt * 16 + l16];
      acc[nt] = __builtin_amdgcn_wmma_f32_16x16x4_f32(
          false, a, false, b, (short)0, acc[nt], false, false);
    }
  }

#pragma unroll
  for (int nt = 0; nt < 4; ++nt) {
#pragma unroll
    for (int v = 0; v < 8; ++v) {
      int m = row0 + v + 8 * half;
      if (m < N) xw[(size_t)m * DFEAT + nt * 16 + l16] = acc[nt][v];
    }
  }
}

// ---------------- degree count at targets ----------------
__global__ __launch_bounds__(256) void gcn_degree(const int* __restrict__ ei,
                                                  float* __restrict__ deg, int E) {
  int e = blockIdx.x * 256 + threadIdx.x;
  if (e < E) atomicAdd(&deg[ei[(size_t)E + e]], 1.0f);
}

__global__ __launch_bounds__(256) void gcn_dinv(const float* __restrict__ deg,
                                                float* __restrict__ dinv, int N) {
  int i = blockIdx.x * 256 + threadIdx.x;
  if (i < N) dinv[i] = 1.0f / sqrtf(deg[i]);   // deg >= 1 guaranteed
}

// ---------------- edge scatter-add: one wave per edge, float2 per lane ----------------
__global__ __launch_bounds__(256) void gcn_aggregate(const int* __restrict__ ei,
                                                     const float* __restrict__ xw,
                                                     const float* __restrict__ dinv,
                                                     float* __restrict__ agg, int E) {
  unsigned t = blockIdx.x * 256u + threadIdx.x;
  unsigned e = t >> 5;
  if (e >= (unsigned)E) return;
  int d = (int)(t & 31u) * 2;
  int r = ei[e];
  int c = ei[(size_t)E + e];
  float norm = dinv[r] * dinv[c];
  const float2* __restrict__ xp =
      reinterpret_cast<const float2*>(xw + (size_t)r * DFEAT);
  float2 v = xp[d >> 1];
  float* dst = agg + (size_t)c * DFEAT + d;
  atomicAdd(dst, v.x * norm);
  atomicAdd(dst + 1, v.y * norm);
}

// ---------------- threefry2x32 (JAX-compatible, 20 rounds) ----------------
__device__ __forceinline__ unsigned rotl32(unsigned x, int r) {
  return (x << r) | (x >> (32 - r));
}

__device__ __forceinline__ void threefry2x32(unsigned k0, unsigned k1,
                                             unsigned& x0, unsigned& x1) {
  unsigned ks[3] = {k0, k1, 0x1BD11BDAu ^ k0 ^ k1};
  x0 += ks[0];
  x1 += ks[1];
  const int rotA[4] = {13, 15, 26, 6};
  const int rotB[4] = {17, 29, 16, 24};
#pragma unroll
  for (int g = 0; g < 5; ++g) {
    const int* rot = (g & 1) ? rotB : rotA;
#pragma unroll
    for (int i = 0; i < 4; ++i) {
      x0 += x1;
      x1 = rotl32(x1, rot[i]);
      x1 ^= x0;
    }
    x0 += ks[(g + 1) % 3];
    x1 += ks[(g + 2) % 3] + (unsigned)(g + 1);
  }
}

// ---------------- finalize: self-loop term + bias, tanh, dropout ----------------
__global__ __launch_bounds__(256) void gcn_finalize(float* __restrict__ out,
                                                    const float* __restrict__ xw,
                                                    const float* __restrict__ dinv,
                                                    const float* __restrict__ bias,
                                                    int N, unsigned halfcount) {
  unsigned j = blockIdx.x * 256u + threadIdx.x;
  unsigned total = (unsigned)N * DFEAT;
  if (j >= total) return;
  unsigned i = j >> 6;
  unsigned d = j & 63u;
  float di = dinv[i];
  float h = tanhf(out[j] + xw[j] * di * di + bias[d]);

  // JAX: bits = threefry2x32(key=(0,42), iota split in halves); u = f32(bits>>9|1.0f)-1
  unsigned x0, x1;
  bool lo = j < halfcount;
  if (lo) { x0 = j; x1 = j + halfcount; }
  else    { x0 = j - halfcount; x1 = j; }
  threefry2x32(0u, 42u, x0, x1);
  unsigned bits = lo ? x0 : x1;
  float u = __uint_as_float((bits >> 9) | 0x3f800000u) - 1.0f;
  out[j] = (u < 0.9f) ? h * (1.0f / 0.9f) : 0.0f;
}

extern "C" void kernel_launch(void* const* d_in, const int* in_sizes, int n_in,
                              void* d_out, int out_size, void* d_ws, size_t ws_size,
                              hipStream_t stream) {
  const float* x  = (const float*)d_in[0];
  const float* W  = (const float*)d_in[1];
  const float* b  = (const float*)d_in[2];
  const int*   ei = (const int*)d_in[3];
  const int N = in_sizes[0] / DFEAT;
  const int E = in_sizes[3] / 2;

  float* out  = (float*)d_out;
  float* xw   = (float*)d_ws;                    // N*64 floats
  float* dinv = xw + (size_t)N * DFEAT;          // N floats
  float* deg  = dinv + N;                        // N floats

  int initBlocks = (N * DFEAT + 255) / 256;
  gcn_init<<<initBlocks, 256, 0, stream>>>(out, deg, N);

  int tiles = (N + 15) / 16;
  gcn_gemm_wmma<<<(tiles + 7) / 8, 256, 0, stream>>>(x, W, xw, N);

  gcn_degree<<<(E + 255) / 256, 256, 0, stream>>>(ei, deg, E);
  gcn_dinv<<<(N + 255) / 256, 256, 0, stream>>>(deg, dinv, N);

  long long aggThreads = (long long)E * 32;
  gcn_aggregate<<<(unsigned)((aggThreads + 255) / 256), 256, 0, stream>>>(
      ei, xw, dinv, out, E);

  unsigned total = (unsigned)N * DFEAT;
  gcn_finalize<<<(total + 255) / 256, 256, 0, stream>>>(out, xw, dinv, b, N,
                                                        total / 2);
}